// SimpleGenerator_39101382262914
// MI455X (gfx1250) — compile-verified
//
#include <hip/hip_runtime.h>
#include <hip/hip_bf16.h>
#include <stdint.h>

// ---------------- problem constants (from the reference) ----------------
#define BB       32          // batch (sentences)
#define BEAM     5
#define VV       128000
#define LL       512
#define PAD_IDX  0
#define EOS_IDX  2
#define K2       10          // 2*BEAM candidates kept per sentence
#define SLICES   8           // slices per (sentence,beam) row
#define SL       (VV / SLICES)   // 16000 words per slice -> 64 KB f32 tile
#define T1       256         // threads, kernel 1
#define T2       128         // threads, kernel 2
#define CAND     (BEAM * SLICES * K2)  // 400 partial candidates per sentence

#define NEG_BIG  (-3.402823466e38f)

typedef unsigned int u32;
typedef u32 v4u __attribute__((ext_vector_type(4)));
typedef int  v4i __attribute__((ext_vector_type(4)));
typedef int  v8i __attribute__((ext_vector_type(8)));

// ---------------------------------------------------------------------------
// TDM: DMA a 1-D run of `nelem` f32 from global memory into LDS.
// Descriptor per CDNA5 ISA ch.7/8 (async_tensor): group0 = count/lds/global/type,
// group1 = data_size + tensor_dim0/tile_dim0 (1 row). Groups 2/3 zero (<=2D).
// clang-23 builtin form (6 args): (v4u, v8i, v4i, v4i, v8i, i32 cpol).
// ---------------------------------------------------------------------------
__device__ __forceinline__ void tdm_load_f32_1d(const float* gsrc,
                                                u32 lds_byte_addr,
                                                u32 nelem) {
  unsigned long long ga = (unsigned long long)(size_t)gsrc;
  v4u g0;
  g0.x = 1u;                                   // count=1 (valid), user mode
  g0.y = lds_byte_addr;                        // lds_addr [63:32]
  g0.z = (u32)(ga & 0xffffffffu);              // global_addr [95:64]
  g0.w = (u32)((ga >> 32) & 0x01ffffffu)       // global_addr [120:96]
       | (2u << 30);                           // type=2 ("image") [127:126]
  v8i g1;
  g1[0] = (int)(2u << 16);                     // wg_mask=0, data_size=2 (4B)
  g1[1] = (int)((nelem & 0xffffu) << 16);      // tensor_dim0[15:0] @bits[63:48]
  g1[2] = (int)(((nelem >> 16) & 0xffffu)      // tensor_dim0[31:16]
       | (1u << 16));                          // tensor_dim1 = 1
  g1[3] = (int)((nelem & 0xffffu) << 16);      // tile_dim0 @bits[127:112]
  g1[4] = 1;                                   // tile_dim1=1, tile_dim2=0
  g1[5] = (int)nelem;                          // tensor_dim0_stride[31:0]
  g1[6] = 0;                                   // stride hi / dim1_stride lo
  g1[7] = 0;
  v4i gz  = {0, 0, 0, 0};
  v8i gz8 = {0, 0, 0, 0, 0, 0, 0, 0};
  __builtin_amdgcn_tensor_load_to_lds(g0, g1, gz, gz, gz8, 0);
}

// ---------------------------------------------------------------------------
// Kernel 1: per-(sentence,beam,slice) top-10.
// TDM streams the 16000-word slice into LDS once (the single HBM pass over
// the 82 MB of scores), then 10 iterative block arg-max rounds run from LDS.
// ---------------------------------------------------------------------------
__global__ __launch_bounds__(T1)
void partial_topk(const float* __restrict__ scores,
                  const float* __restrict__ beam_scores,
                  float* __restrict__ part_s,
                  int*   __restrict__ part_i) {
  __shared__ float tile[SL];          // 64000 B
  __shared__ float red_s[T1];
  __shared__ int   red_i[T1];

  const int blk = blockIdx.x;                 // (b*BEAM + m)*SLICES + s
  const int s   = blk % SLICES;
  const int m   = (blk / SLICES) % BEAM;
  const int b   = blk / (SLICES * BEAM);
  const int tid = threadIdx.x;

  const float* src = scores + ((size_t)(b * BEAM + m) * VV + (size_t)s * SL);

  if (tid < 32u) {                            // wave 0 drives the TDM
    tdm_load_f32_1d(src, (u32)(size_t)(void*)tile, (u32)SL);
    __builtin_amdgcn_s_wait_tensorcnt(0);     // TENSORcnt == 0
  }
  __syncthreads();

  const float bs = beam_scores[b * BEAM + m]; // uniform scalar add (post-hoc)

  for (int k = 0; k < K2; ++k) {
    // local arg-max over strided LDS elements (strict > : earliest index wins)
    float best = NEG_BIG;
    int   bi   = 0;
    for (int i = tid; i < SL; i += T1) {
      float v = tile[i];
      if (v > best) { best = v; bi = i; }
    }
    red_s[tid] = best;
    red_i[tid] = bi;
    __syncthreads();
    for (int off = T1 >> 1; off > 0; off >>= 1) {
      if (tid < off) {
        float vo = red_s[tid + off];
        if (vo > red_s[tid]) { red_s[tid] = vo; red_i[tid] = red_i[tid + off]; }
      }
      __syncthreads();
    }
    if (tid == 0) {
      const int w = red_i[0];
      part_s[(size_t)blk * K2 + k] = red_s[0] + bs;
      part_i[(size_t)blk * K2 + k] = m * VV + (s * SL + w);  // candidate id
      tile[w] = NEG_BIG;                                      // mask winner
    }
    __syncthreads();
  }
}

// ---------------------------------------------------------------------------
// Kernel 2: per-sentence merge of 400 partials -> top-10 (descending), then
// the reference's EOS filter: first 5 non-EOS in rank order; fallback slots
// get (score=0, word=PAD, beam-row 0).
// ---------------------------------------------------------------------------
__global__ __launch_bounds__(T2)
void final_topk(const float* __restrict__ part_s,
                const int*   __restrict__ part_i,
                float* __restrict__ out_scores,   // d_out tail [B*BEAM]
                int*   __restrict__ sel_words,    // ws [B*BEAM]
                int*   __restrict__ eff_beam) {   // ws [B*BEAM]
  __shared__ float cs[CAND];
  __shared__ int   ci[CAND];
  __shared__ float red_s[T2];
  __shared__ int   red_i[T2];
  __shared__ float top_s[K2];
  __shared__ int   top_i[K2];

  const int b   = blockIdx.x;
  const int tid = threadIdx.x;

  for (int i = tid; i < CAND; i += T2) {
    cs[i] = part_s[(size_t)b * CAND + i];
    ci[i] = part_i[(size_t)b * CAND + i];
  }
  __syncthreads();

  for (int k = 0; k < K2; ++k) {
    float best = NEG_BIG;
    int   bp   = 0;
    for (int i = tid; i < CAND; i += T2) {
      float v = cs[i];
      if (v > best) { best = v; bp = i; }
    }
    red_s[tid] = best;
    red_i[tid] = bp;
    __syncthreads();
    for (int off = T2 >> 1; off > 0; off >>= 1) {
      if (tid < off) {
        float vo = red_s[tid + off];
        if (vo > red_s[tid]) { red_s[tid] = vo; red_i[tid] = red_i[tid + off]; }
      }
      __syncthreads();
    }
    if (tid == 0) {
      const int p = red_i[0];
      top_s[k] = red_s[0];
      top_i[k] = ci[p];
      cs[p]    = NEG_BIG;
    }
    __syncthreads();
  }

  if (tid == 0) {
    float ss[BEAM];
    int   sw[BEAM], eb[BEAM];
    for (int q = 0; q < BEAM; ++q) { ss[q] = 0.0f; sw[q] = PAD_IDX; eb[q] = 0; }
    int cnt = 0;
    for (int k = 0; k < K2 && cnt < BEAM; ++k) {
      const int c  = top_i[k];
      const int w  = c % VV;
      const int mm = c / VV;
      if (w != EOS_IDX) {
        ss[cnt] = top_s[k];
        sw[cnt] = w;
        eb[cnt] = b * BEAM + mm;
        ++cnt;
      }
    }
    for (int q = 0; q < BEAM; ++q) {
      out_scores[b * BEAM + q] = ss[q];
      sel_words [b * BEAM + q] = sw[q];
      eff_beam  [b * BEAM + q] = eb[q];
    }
  }
}

// ---------------------------------------------------------------------------
// Kernel 3: reorder hypotheses and append the new word at column cur_len.
// ---------------------------------------------------------------------------
__global__ __launch_bounds__(128)
void regen(const int* __restrict__ generated,
           const int* __restrict__ sel_words,
           const int* __restrict__ eff_beam,
           const int* __restrict__ cur_len_p,
           int* __restrict__ out_gen) {
  const int r  = blockIdx.x;            // 0 .. B*BEAM-1
  const int cl = cur_len_p[0];
  const int sr = eff_beam[r];
  const int w  = sel_words[r];
  const int* src = generated + (size_t)sr * LL;
  int*       dst = out_gen   + (size_t)r  * LL;
  for (int i = threadIdx.x; i < LL; i += blockDim.x) {
    int v = src[i];
    dst[i] = (i == cl) ? w : v;
  }
}

// ---------------------------------------------------------------------------
extern "C" void kernel_launch(void* const* d_in, const int* in_sizes, int n_in,
                              void* d_out, int out_size, void* d_ws, size_t ws_size,
                              hipStream_t stream) {
  (void)in_sizes; (void)n_in; (void)out_size; (void)ws_size;

  const float* scores      = (const float*)d_in[0];   // [B*BEAM, V] f32
  const float* beam_scores = (const float*)d_in[1];   // [B, BEAM]  f32
  const int*   generated   = (const int*)  d_in[2];   // [B*BEAM, L] i32
  const int*   cur_len     = (const int*)  d_in[3];   // scalar i32

  int*   out_gen = (int*)d_out;                               // 81920 i32
  float* out_sc  = (float*)d_out + (size_t)BB * BEAM * LL;    // 160 f32

  // workspace carve-up
  const int nblk1 = BB * BEAM * SLICES;           // 1280
  float* part_s    = (float*)d_ws;                            // 12800 f32
  int*   part_i    = (int*)((char*)d_ws + (size_t)nblk1 * K2 * 4);
  int*   sel_words = part_i + (size_t)nblk1 * K2;
  int*   eff_beam  = sel_words + BB * BEAM;

  partial_topk<<<nblk1, T1, 0, stream>>>(scores, beam_scores, part_s, part_i);
  final_topk  <<<BB,    T2, 0, stream>>>(part_s, part_i, out_sc, sel_words, eff_beam);
  regen       <<<BB * BEAM, 128, 0, stream>>>(generated, sel_words, eff_beam,
                                              cur_len, out_gen);
}